// VoxelTransformer_82248623719069
// MI455X (gfx1250) — compile-verified
//
#include <hip/hip_runtime.h>
#include <hip/hip_bf16.h>
#include <math.h>
#include <stdint.h>

// ---------------- problem constants ----------------
#define BATCH 2
#define NPTS  8192
#define MPTS  4096      // N/2
#define KNN_K 3
#define FD    64
#define SD    5
#define AD    256
#define R2    (BATCH*MPTS)        // 8192 rows after pooling
#define RS    (R2*KNN_K)          // 24576 rows into SA gemm
#define KSA   72                  // 69 padded to multiple of 4

typedef float v2f __attribute__((ext_vector_type(2)));
typedef float v8f __attribute__((ext_vector_type(8)));

// ---- CDNA5 async copy (global -> LDS, ASYNCcnt-tracked) ----
__device__ __forceinline__ void async_b128(unsigned lds_byte_addr, const void* gaddr) {
    asm volatile("global_load_async_to_lds_b128 %0, %1, off"
                 :: "v"(lds_byte_addr), "v"(gaddr)
                 : "memory");
}
__device__ __forceinline__ void async_wait0() {
    asm volatile("s_wait_asynccnt 0" ::: "memory");
}

// =====================================================================
// 1) Farthest point sampling: one block per batch, dist[] in LDS.
//    Matches jnp.argmax first-occurrence tie-breaking.
// =====================================================================
#define FPS_T 1024
__global__ void fps_kernel(const float* __restrict__ spatial, int* __restrict__ idxs) {
    __shared__ float dist[NPTS];          // 32 KB
    __shared__ float rv[FPS_T];
    __shared__ int   ri[FPS_T];
    __shared__ float lastp[SD];
    const int b   = blockIdx.x;
    const int tid = threadIdx.x;
    const float* sp = spatial + (size_t)b * NPTS * SD;

    for (int j = tid; j < NPTS; j += FPS_T) dist[j] = INFINITY;
    if (tid == 0) {
        dist[0] = -INFINITY;              // mask0[0] = True
        idxs[b * MPTS] = 0;
        for (int c = 0; c < SD; ++c) lastp[c] = sp[c];
    }
    __syncthreads();

    for (int i = 1; i < MPTS; ++i) {
        const float l0 = lastp[0], l1 = lastp[1], l2 = lastp[2], l3 = lastp[3], l4 = lastp[4];
        float bv = -INFINITY; int bi = 0;
        for (int j = tid; j < NPTS; j += FPS_T) {
            const float* p = sp + j * SD;
            float d0 = p[0]-l0, d1 = p[1]-l1, d2 = p[2]-l2, d3 = p[3]-l3, d4 = p[4]-l4;
            float d  = d0*d0 + d1*d1 + d2*d2 + d3*d3 + d4*d4;
            float nd = fminf(dist[j], d); // -inf stays -inf for masked entries
            dist[j]  = nd;
            if (nd > bv) { bv = nd; bi = j; }   // ascending j -> first max
        }
        rv[tid] = bv; ri[tid] = bi;
        __syncthreads();
        for (int s = FPS_T / 2; s > 0; s >>= 1) {
            if (tid < s) {
                float ov = rv[tid + s]; int oi = ri[tid + s];
                if (ov > rv[tid] || (ov == rv[tid] && oi < ri[tid])) { rv[tid] = ov; ri[tid] = oi; }
            }
            __syncthreads();
        }
        if (tid == 0) {
            int sel = ri[0];
            idxs[b * MPTS + i] = sel;
            dist[sel] = -INFINITY;        // mask.at[sel].set(True)
            const float* p = sp + sel * SD;
            for (int c = 0; c < SD; ++c) lastp[c] = p[c];
        }
        __syncthreads();
    }
}

// =====================================================================
// 2) Brute-force KNN (top-3 nearest, self excluded) + centroid output.
// =====================================================================
#define KNN_CH 512
__global__ void knn_kernel(const float* __restrict__ spatial, const int* __restrict__ idxs,
                           int* __restrict__ nn, float* __restrict__ cent_out) {
    __shared__ float chp[KNN_CH * SD];    // 10 KB
    const int tid = threadIdx.x;
    const int cid = blockIdx.x * 256 + tid;       // [0, R2)
    const int b   = cid >> 12;                    // /4096
    const float* sp = spatial + (size_t)b * NPTS * SD;
    const int ci = idxs[cid];
    float c0, c1, c2, c3, c4;
    {
        const float* p = sp + ci * SD;
        c0 = p[0]; c1 = p[1]; c2 = p[2]; c3 = p[3]; c4 = p[4];
    }
    float* co = cent_out + (size_t)cid * SD;
    co[0] = c0; co[1] = c1; co[2] = c2; co[3] = c3; co[4] = c4;

    float dd0 = INFINITY, dd1 = INFINITY, dd2 = INFINITY;
    int   ii0 = 0, ii1 = 0, ii2 = 0;

    for (int ch = 0; ch < NPTS / KNN_CH; ++ch) {
        __syncthreads();
        for (int t = tid; t < KNN_CH * SD; t += 256) chp[t] = sp[ch * KNN_CH * SD + t];
        __syncthreads();
        for (int p = 0; p < KNN_CH; ++p) {
            int j = ch * KNN_CH + p;
            const float* q = chp + p * SD;
            float e0 = q[0]-c0, e1 = q[1]-c1, e2 = q[2]-c2, e3 = q[3]-c3, e4 = q[4]-c4;
            float d2v = e0*e0 + e1*e1 + e2*e2 + e3*e3 + e4*e4;
            if (j != ci && d2v < dd2) {           // strict < : equal dist keeps earlier j
                if (d2v < dd1) {
                    dd2 = dd1; ii2 = ii1;
                    if (d2v < dd0) { dd1 = dd0; ii1 = ii0; dd0 = d2v; ii0 = j; }
                    else           { dd1 = d2v; ii1 = j; }
                } else { dd2 = d2v; ii2 = j; }
            }
        }
    }
    nn[cid * KNN_K + 0] = ii0;
    nn[cid * KNN_K + 1] = ii1;
    nn[cid * KNN_K + 2] = ii2;
}

// =====================================================================
// 3) Pad W_sa [69x256] -> [72x256] (zero rows 69..71)
// =====================================================================
__global__ void pad_wsa_kernel(const float* __restrict__ W, float* __restrict__ Wp) {
    int t = blockIdx.x * 256 + threadIdx.x;       // 72*256
    int r = t >> 8, c = t & 255;
    Wp[t] = (r < FD + SD) ? W[r * AD + c] : 0.0f;
}

// =====================================================================
// 4) Build SA input matrix A_sa [RS x 72] = [nbr_feat(64) | rel(5) | 0 pad]
// =====================================================================
__global__ void build_asa_kernel(const float* __restrict__ x, const float* __restrict__ spatial,
                                 const int* __restrict__ idxs, const int* __restrict__ nn,
                                 float* __restrict__ Asa) {
    int t = blockIdx.x * 256 + threadIdx.x;
    if (t >= RS * KSA) return;
    int row = t / KSA, col = t - row * KSA;
    int b   = row / (MPTS * KNN_K);
    int rem = row - b * (MPTS * KNN_K);
    int m   = rem / KNN_K;
    int j   = nn[row];
    float val;
    if (col < FD) {
        val = x[((size_t)b * NPTS + j) * FD + col];
    } else if (col < FD + SD) {
        int c5 = col - FD;
        int ci = idxs[b * MPTS + m];
        val = spatial[((size_t)b * NPTS + j) * SD + c5]
            - spatial[((size_t)b * NPTS + ci) * SD + c5];
    } else {
        val = 0.0f;
    }
    Asa[t] = val;
}

// =====================================================================
// 5) f32 WMMA GEMM:  C[R x 256] = epi( preA(A,A2) @ W + bias )
//    32-row tiles: 2 M-subtiles per wave share each B fragment (2x reuse).
//    A tile staged via GLOBAL_LOAD_ASYNC_TO_LDS_B128 when no A-transform.
//    LDA = KD+4 floats: rows stay 16B-aligned AND lane stride = 4 banks.
//    AOP: 0 A, 1 A-A2, 2 A*A2.  EPI: 0 bias, 1 bias+relu, 2 bias+RES.
// =====================================================================
template <int KD, int AOP, int EPI>
__global__ void gemm32t(const float* __restrict__ A, const float* __restrict__ A2,
                        const float* __restrict__ W, const float* __restrict__ bias,
                        const float* __restrict__ RES, float* __restrict__ C) {
    constexpr int NO  = AD;
    constexpr int LDA = KD + 4;
    __shared__ float As[32 * LDA];       // KD=256: 33.3 KB, KD=72: 9.7 KB
    const int tid   = threadIdx.x;
    const int wave  = tid >> 5;
    const int lane  = tid & 31;
    const int tileM = blockIdx.x * 32;
    const int tileN = (blockIdx.y * 8 + wave) * 16;

    // prefetch the W panel this wave will stream (L2 warm-up)
    for (int r = lane; r < KD; r += 32)
        __builtin_prefetch(W + (size_t)r * NO + tileN, 0, 3);

    if (AOP == 0) {
        constexpr int CPR = KD / 4;       // 16-byte chunks per row
        for (int t = tid; t < 32 * CPR; t += 256) {
            int r = t / CPR, c = t - r * CPR;
            unsigned lds = (unsigned)(uintptr_t)&As[r * LDA + c * 4];
            async_b128(lds, A + (size_t)(tileM + r) * KD + c * 4);
        }
        async_wait0();                    // this wave's async copies done
    } else {
        for (int t = tid; t < 32 * KD; t += 256) {
            int r = t / KD, c = t - r * KD;
            size_t gi = (size_t)(tileM + r) * KD + c;
            float v = A[gi];
            if (AOP == 1) v -= A2[gi];
            if (AOP == 2) v *= A2[gi];
            As[r * LDA + c] = v;
        }
    }
    __syncthreads();

    v8f acc0 = {0.f, 0.f, 0.f, 0.f, 0.f, 0.f, 0.f, 0.f};
    v8f acc1 = {0.f, 0.f, 0.f, 0.f, 0.f, 0.f, 0.f, 0.f};
    const int lm   = lane & 15;
    const int koff = (lane >> 4) << 1;    // VGPR0 -> K{0,2}, VGPR1 -> K{1,3}
    for (int k0 = 0; k0 < KD; k0 += 4) {
        const float* wp = W + (size_t)(k0 + koff) * NO + tileN + lm;
        v2f bb;
        bb.x = wp[0];
        bb.y = wp[NO];
        v2f a0, a1;
        a0.x = As[lm * LDA + k0 + koff];
        a0.y = As[lm * LDA + k0 + koff + 1];
        a1.x = As[(16 + lm) * LDA + k0 + koff];
        a1.y = As[(16 + lm) * LDA + k0 + koff + 1];
        acc0 = __builtin_amdgcn_wmma_f32_16x16x4_f32(false, a0, false, bb,
                                                     (short)0, acc0, false, false);
        acc1 = __builtin_amdgcn_wmma_f32_16x16x4_f32(false, a1, false, bb,
                                                     (short)0, acc1, false, false);
    }

    const int rowOff = (lane >> 4) << 3;  // M = r (lanes 0-15) / r+8 (16-31)
    const int col    = tileN + lm;
    const float bcol = bias[col];
#pragma unroll
    for (int s2 = 0; s2 < 2; ++s2) {
        const v8f& acc = s2 ? acc1 : acc0;
#pragma unroll
        for (int r = 0; r < 8; ++r) {
            int row = tileM + s2 * 16 + rowOff + r;
            float v = acc[r] + bcol;
            if (EPI == 1) v = fmaxf(v, 0.0f);
            if (EPI == 2) v += RES[(size_t)row * NO + col];
            C[(size_t)row * NO + col] = v;
        }
    }
}

// =====================================================================
// 6) per-centroid BN scale/shift + ReLU + max over K=3 neighbor rows
// =====================================================================
__global__ void bnmax_kernel(const float* __restrict__ h, const float* __restrict__ gamma,
                             const float* __restrict__ beta, const float* __restrict__ mean,
                             const float* __restrict__ var, float* __restrict__ f) {
    int t = blockIdx.x * 256 + threadIdx.x;       // R2*256
    int r = t >> 8;
    int m = r & (MPTS - 1);                       // row = b*M + m -> BN indexed by m
    float inv = gamma[m] * rsqrtf(var[m] + 1e-5f);
    float sh  = beta[m] - mean[m] * inv;
    const float* hp = h + (size_t)(r * KNN_K) * AD + (t & 255);
    float v0 = fmaxf(hp[0]      * inv + sh, 0.0f);
    float v1 = fmaxf(hp[AD]     * inv + sh, 0.0f);
    float v2 = fmaxf(hp[2 * AD] * inv + sh, 0.0f);
    f[t] = fmaxf(v0, fmaxf(v1, v2));
}

// =====================================================================
// 7) row softmax over 256 (in place), one block per row
// =====================================================================
__global__ void softmax256_kernel(float* __restrict__ s) {
    __shared__ float red[AD];
    const int r = blockIdx.x, c = threadIdx.x;
    float v = s[(size_t)r * AD + c];
    red[c] = v; __syncthreads();
    for (int st = 128; st > 0; st >>= 1) {
        if (c < st) red[c] = fmaxf(red[c], red[c + st]);
        __syncthreads();
    }
    float mx = red[0];
    __syncthreads();
    float e = expf(v - mx);
    red[c] = e; __syncthreads();
    for (int st = 128; st > 0; st >>= 1) {
        if (c < st) red[c] += red[c + st];
        __syncthreads();
    }
    s[(size_t)r * AD + c] = e / red[0];
}

// =====================================================================
// launch
// =====================================================================
extern "C" void kernel_launch(void* const* d_in, const int* in_sizes, int n_in,
                              void* d_out, int out_size, void* d_ws, size_t ws_size,
                              hipStream_t stream) {
    const float* x       = (const float*)d_in[0];
    const float* spatial = (const float*)d_in[1];
    const float* W_sa    = (const float*)d_in[2];
    const float* b_sa    = (const float*)d_in[3];
    const float* gamma   = (const float*)d_in[4];
    const float* beta    = (const float*)d_in[5];
    const float* mean    = (const float*)d_in[6];
    const float* var     = (const float*)d_in[7];
    const float* W_in    = (const float*)d_in[8];
    const float* b_in    = (const float*)d_in[9];
    const float* W_q     = (const float*)d_in[10];
    const float* b_q     = (const float*)d_in[11];
    const float* W_k     = (const float*)d_in[12];
    const float* b_k     = (const float*)d_in[13];
    const float* W_v     = (const float*)d_in[14];
    const float* b_v     = (const float*)d_in[15];
    const float* W_w1    = (const float*)d_in[16];
    const float* b_w1    = (const float*)d_in[17];
    const float* W_w2    = (const float*)d_in[18];
    const float* b_w2    = (const float*)d_in[19];
    const float* W_out   = (const float*)d_in[20];
    const float* b_out   = (const float*)d_in[21];

    char* ws = (char*)d_ws;
    int*   idxs = (int*)  (ws + 0);                       //  32 KB
    int*   nn   = (int*)  (ws + 32768);                   //  96 KB
    float* Wsap = (float*)(ws + 131072);                  //  72 KB
    float* Asa  = (float*)(ws + 204800);                  // 7.08 MB
    float* h    = (float*)(ws + 7282688);                 // 25.2 MB
    float* f    = (float*)(ws + 32448512);                // 8.39 MB each below
    float* t    = (float*)(ws + 40837120);
    float* q    = (float*)(ws + 49225728);
    float* kk   = (float*)(ws + 57614336);
    float* v    = (float*)(ws + 66002944);
    float* u    = h;                                      // reuse dead h region
    float* s    = (float*)(ws + 7282688 + 8388608);       // second half of h region

    float* out  = (float*)d_out;                          // [B, M, 256]
    float* cent = out + (size_t)R2 * AD;                  // [B, M, 5]

    // pipeline
    fps_kernel<<<BATCH, FPS_T, 0, stream>>>(spatial, idxs);
    knn_kernel<<<R2 / 256, 256, 0, stream>>>(spatial, idxs, nn, cent);
    pad_wsa_kernel<<<(KSA * AD) / 256, 256, 0, stream>>>(W_sa, Wsap);
    build_asa_kernel<<<(RS * KSA + 255) / 256, 256, 0, stream>>>(x, spatial, idxs, nn, Asa);

    gemm32t<KSA, 0, 0><<<dim3(RS / 32, 2), 256, 0, stream>>>(Asa, nullptr, Wsap, b_sa, nullptr, h);
    bnmax_kernel<<<(R2 * AD) / 256, 256, 0, stream>>>(h, gamma, beta, mean, var, f);

    dim3 g(R2 / 32, 2);
    gemm32t<AD, 0, 0><<<g, 256, 0, stream>>>(f, nullptr, W_in, b_in, nullptr, t);
    gemm32t<AD, 0, 0><<<g, 256, 0, stream>>>(t, nullptr, W_q,  b_q,  nullptr, q);
    gemm32t<AD, 0, 0><<<g, 256, 0, stream>>>(t, nullptr, W_k,  b_k,  nullptr, kk);
    gemm32t<AD, 0, 0><<<g, 256, 0, stream>>>(t, nullptr, W_v,  b_v,  nullptr, v);
    gemm32t<AD, 1, 1><<<g, 256, 0, stream>>>(q, kk,      W_w1, b_w1, nullptr, u);   // relu(q-k @ W1)
    gemm32t<AD, 0, 0><<<g, 256, 0, stream>>>(u, nullptr, W_w2, b_w2, nullptr, s);
    softmax256_kernel<<<R2, AD, 0, stream>>>(s);
    gemm32t<AD, 2, 2><<<g, 256, 0, stream>>>(s, v,       W_out, b_out, f, out);     // (w*v)@Wo + b + f
}